// Smoothing_19000935317762
// MI455X (gfx1250) — compile-verified
//
#include <hip/hip_runtime.h>
#include <stdint.h>

// ============================================================================
// 3D Gaussian smoothing (9x9x9, reflect pad) + clip-to-band, 256^3 f32.
// Separable: three 1D passes (X, Y, Z), Z-pass fused with the clip.
// Memory-bound (~450 MiB of traffic, ~1 GFLOP): optimize for streaming.
// CDNA5 path: async global->LDS (ASYNCcnt) for the X-pass halo staging.
// ============================================================================

#define GN   256
#define PLANE (GN * GN)          // 65536
#define NROW  (GN * GN)          // rows for X pass
#define PADK 4
#define CLIPV 0.03f

#define AS1 __attribute__((address_space(1)))
#define AS3 __attribute__((address_space(3)))

typedef int v4i_t __attribute__((ext_vector_type(4)));

#if defined(__has_builtin)
# if __has_builtin(__builtin_amdgcn_global_load_async_to_lds_b32)
#  define HAVE_ASYNC_B32 1
# endif
# if __has_builtin(__builtin_amdgcn_global_load_async_to_lds_b128)
#  define HAVE_ASYNC_B128 1
# endif
# if __has_builtin(__builtin_amdgcn_s_wait_asynccnt)
#  define HAVE_WAIT_ASYNC 1
# endif
#endif

__device__ __forceinline__ int reflect_i(int i) {
    i = (i < 0) ? -i : i;
    i = (i > GN - 1) ? (2 * (GN - 1) - i) : i;
    return i;
}

// --- async global -> LDS copies (per-lane gather into LDS) -------------------
__device__ __forceinline__ void async_g2l_b32(const float* g, float* l) {
#if defined(HAVE_ASYNC_B32)
    __builtin_amdgcn_global_load_async_to_lds_b32(
        (AS1 int*)(uintptr_t)g,
        (AS3 int*)(uint32_t)(uintptr_t)l, 0, 0);
#else
    asm volatile("global_load_async_to_lds_b32 %0, %1, off"
                 :: "v"((uint32_t)(uintptr_t)l), "v"(g) : "memory");
#endif
}

__device__ __forceinline__ void async_g2l_b128(const float* g, float* l) {
#if defined(HAVE_ASYNC_B128)
    __builtin_amdgcn_global_load_async_to_lds_b128(
        (AS1 v4i_t*)(uintptr_t)g,
        (AS3 v4i_t*)(uint32_t)(uintptr_t)l, 0, 0);
#else
    asm volatile("global_load_async_to_lds_b128 %0, %1, off"
                 :: "v"((uint32_t)(uintptr_t)l), "v"(g) : "memory");
#endif
}

__device__ __forceinline__ void wait_async0() {
#if defined(HAVE_WAIT_ASYNC)
    __builtin_amdgcn_s_wait_asynccnt(0);
#else
    asm volatile("s_wait_asynccnt 0" ::: "memory");
#endif
    asm volatile("" ::: "memory");   // keep LDS reads below the wait
}

// --- small float4 helpers ----------------------------------------------------
__device__ __forceinline__ float4 mul4(float wk, float4 a) {
    return make_float4(wk * a.x, wk * a.y, wk * a.z, wk * a.w);
}
__device__ __forceinline__ float4 fma4(float wk, float4 a, float4 s) {
    s.x = fmaf(wk, a.x, s.x); s.y = fmaf(wk, a.y, s.y);
    s.z = fmaf(wk, a.z, s.z); s.w = fmaf(wk, a.w, s.w);
    return s;
}

// ----------------------------------------------------------------------------
// 1D weights: w[a] = sum_{b,c} k3[a,b,c]  (k3 is the normalized 3D kernel, so
// the marginal over two axes is exactly g/sum(g)).
// Fast path: weights precomputed once into d_ws tail -> 9 uniform s_loads.
// Fallback (wpre == nullptr): in-block marginal, fully unrolled so the 81
// loads issue as one clause (one memory round-trip, not 81).
// ----------------------------------------------------------------------------
__device__ __forceinline__ void get_weights(const float* __restrict__ k3,
                                            const float* __restrict__ wpre,
                                            float* __restrict__ ws9,
                                            float w[9], int tid) {
    if (wpre != nullptr) {
        #pragma unroll
        for (int k = 0; k < 9; ++k) w[k] = wpre[k];
        return;
    }
    if (tid < 9) {
        float s = 0.f;
        #pragma unroll
        for (int i = 0; i < 81; ++i) s += k3[tid * 81 + i];
        ws9[tid] = s;
    }
    __syncthreads();
    #pragma unroll
    for (int k = 0; k < 9; ++k) w[k] = ws9[k];
}

// One-shot weight precompute (1 block, wave-sized).
__global__ void wprep_kernel(const float* __restrict__ k3, float* __restrict__ w9) {
    const int a = threadIdx.x;
    if (a < 9) {
        float s = 0.f;
        #pragma unroll
        for (int i = 0; i < 81; ++i) s += k3[a * 81 + i];
        w9[a] = s;
    }
}

// ============================================================================
// Pass 1: convolve along X (contiguous axis).
// One wave per row (8 rows / 256-thread block). Each wave async-stages its
// 264-float padded row into a wave-private LDS region (no workgroup barrier
// needed on the fast path), then each lane computes 8 contiguous outputs from
// a 16-float window (ds_load_b128 in, global_store_b128 out).
// ============================================================================
__global__ __launch_bounds__(256) void convx_kernel(const float* __restrict__ src,
                                                    float* __restrict__ dst,
                                                    const float* __restrict__ k3,
                                                    const float* __restrict__ wpre) {
    __shared__ __align__(16) float tile[8][264];
    __shared__ float ws9[9];

    const int tid  = threadIdx.x;
    const int wid  = tid >> 5;
    const int lane = tid & 31;

    float w[9];
    get_weights(k3, wpre, ws9, w, tid);

    const int row = blockIdx.x * 8 + wid;
    const float* __restrict__ srow = src + (size_t)row * GN;
    float* trow = &tile[wid][0];

    // Bulk interior: tile positions p = 4..259 map linearly to gx = p-4 (0..255).
    #pragma unroll
    for (int j = 0; j < 2; ++j) {
        const int p = 4 + lane * 4 + 128 * j;       // 16B-aligned both sides
        async_g2l_b128(srow + (p - 4), trow + p);
    }
    // Reflect halos (reversed order -> scalar b32 gathers): p=0..3 and 260..263.
    if (lane < 8) {
        const int p = (lane < 4) ? lane : (256 + lane);
        async_g2l_b32(srow + reflect_i(p - PADK), trow + p);
    }
    wait_async0();

    const int x0 = lane * 8;
    const float4* lp = (const float4*)(trow + x0);   // 16B aligned
    const float4 v0 = lp[0], v1 = lp[1], v2 = lp[2], v3 = lp[3];
    float f[16] = { v0.x, v0.y, v0.z, v0.w,  v1.x, v1.y, v1.z, v1.w,
                    v2.x, v2.y, v2.z, v2.w,  v3.x, v3.y, v3.z, v3.w };

    float o[8];
    #pragma unroll
    for (int j = 0; j < 8; ++j) {
        float s = w[0] * f[j];
        #pragma unroll
        for (int k = 1; k < 9; ++k) s = fmaf(w[k], f[j + k], s);
        o[j] = s;
    }

    float4* op = (float4*)(dst + (size_t)row * GN + x0);
    op[0] = make_float4(o[0], o[1], o[2], o[3]);
    op[1] = make_float4(o[4], o[5], o[6], o[7]);
}

// ============================================================================
// Pass 2: convolve along Y. Lanes cover contiguous x as float4 (coalesced
// B128); each thread walks a 32-row y-strip with a 9-deep rolling window:
// 1 load + 36 FMA + 1 store per 4 outputs.
// block = 64 x-quads * 4 z ; grid = 64 z-groups * 8 y-strips = 512 blocks.
// ============================================================================
__global__ __launch_bounds__(256) void convy_kernel(const float* __restrict__ src,
                                                    float* __restrict__ dst,
                                                    const float* __restrict__ k3,
                                                    const float* __restrict__ wpre) {
    __shared__ float ws9[9];
    const int tid = threadIdx.x;
    float w[9];
    get_weights(k3, wpre, ws9, w, tid);

    const int xq = tid & 63;
    const int zl = tid >> 6;
    const int zg = blockIdx.x >> 3;
    const int ys = blockIdx.x & 7;
    const int z  = zg * 4 + zl;
    const int x  = xq * 4;
    const int y0 = ys * 32;

    const float* __restrict__ base  = src + (size_t)z * PLANE + x;
    float* __restrict__       obase = dst + (size_t)z * PLANE + x;

    float4 c[9];
    #pragma unroll
    for (int k = 0; k < 9; ++k)
        c[k] = *(const float4*)(base + (size_t)reflect_i(y0 - PADK + k) * GN);

    #pragma unroll 4
    for (int i = 0; i < 32; ++i) {
        const int y = y0 + i;
        float4 s = mul4(w[0], c[0]);
        #pragma unroll
        for (int k = 1; k < 9; ++k) s = fma4(w[k], c[k], s);
        *(float4*)(obase + (size_t)y * GN) = s;
        #pragma unroll
        for (int k = 0; k < 8; ++k) c[k] = c[k + 1];
        c[8] = *(const float4*)(base + (size_t)reflect_i(y + 5) * GN);
    }
}

// ============================================================================
// Pass 3: convolve along Z (stride 65536) with the same rolling window,
// fused with the final clip: out = clamp(x, smooth - 0.03, smooth + 0.03).
// block = 64 x-quads * 4 y ; grid = 64 y-groups * 8 z-strips = 512 blocks.
// ============================================================================
__global__ __launch_bounds__(256) void convz_clip_kernel(const float* __restrict__ src,
                                                         const float* __restrict__ orig,
                                                         float* __restrict__ dst,
                                                         const float* __restrict__ k3,
                                                         const float* __restrict__ wpre) {
    __shared__ float ws9[9];
    const int tid = threadIdx.x;
    float w[9];
    get_weights(k3, wpre, ws9, w, tid);

    const int xq = tid & 63;
    const int yl = tid >> 6;
    const int yg = blockIdx.x >> 3;
    const int zs = blockIdx.x & 7;
    const int y  = yg * 4 + yl;
    const int x  = xq * 4;
    const int z0 = zs * 32;

    const float* __restrict__ base = src  + (size_t)y * GN + x;
    const float* __restrict__ ob   = orig + (size_t)y * GN + x;
    float* __restrict__       db   = dst  + (size_t)y * GN + x;

    float4 c[9];
    #pragma unroll
    for (int k = 0; k < 9; ++k)
        c[k] = *(const float4*)(base + (size_t)reflect_i(z0 - PADK + k) * PLANE);

    #pragma unroll 4
    for (int i = 0; i < 32; ++i) {
        const int z = z0 + i;
        float4 s = mul4(w[0], c[0]);
        #pragma unroll
        for (int k = 1; k < 9; ++k) s = fma4(w[k], c[k], s);

        const float4 o = *(const float4*)(ob + (size_t)z * PLANE);
        float4 r;
        r.x = fmaxf(fminf(o.x, s.x + CLIPV), s.x - CLIPV);
        r.y = fmaxf(fminf(o.y, s.y + CLIPV), s.y - CLIPV);
        r.z = fmaxf(fminf(o.z, s.z + CLIPV), s.z - CLIPV);
        r.w = fmaxf(fminf(o.w, s.w + CLIPV), s.w - CLIPV);
        *(float4*)(db + (size_t)z * PLANE) = r;

        #pragma unroll
        for (int k = 0; k < 8; ++k) c[k] = c[k + 1];
        c[8] = *(const float4*)(base + (size_t)reflect_i(z + 5) * PLANE);
    }
}

// ============================================================================
// Launch: x --convX--> d_out --convY--> ws(64 MiB) --convZ+clip--> d_out
// d_ws layout: [0, 64 MiB) intermediate volume; [64 MiB, +64 B) 1D weights
// (weights precompute is skipped if ws_size is exactly the volume size; the
// conv kernels then derive weights per block instead).
// ============================================================================
extern "C" void kernel_launch(void* const* d_in, const int* in_sizes, int n_in,
                              void* d_out, int out_size, void* d_ws, size_t ws_size,
                              hipStream_t stream) {
    (void)in_sizes; (void)n_in; (void)out_size;

    const float* x  = (const float*)d_in[0];   // 256^3 f32
    const float* k3 = (const float*)d_in[1];   // 9*9*9 f32 (normalized)
    float* out = (float*)d_out;
    float* tmp = (float*)d_ws;                 // 64 MiB intermediate

    const size_t TMP_BYTES = (size_t)GN * GN * GN * sizeof(float);
    float* w9 = nullptr;
    if (ws_size >= TMP_BYTES + 64) {
        w9 = (float*)((char*)d_ws + TMP_BYTES);
        wprep_kernel<<<1, 32, 0, stream>>>(k3, w9);
    }

    convx_kernel<<<NROW / 8, 256, 0, stream>>>(x, out, k3, w9);
    convy_kernel<<<512, 256, 0, stream>>>(out, tmp, k3, w9);
    convz_clip_kernel<<<512, 256, 0, stream>>>(tmp, x, out, k3, w9);
}